// EdgeCrossingLoss_16166256902862
// MI455X (gfx1250) — compile-verified
//
#include <hip/hip_runtime.h>

typedef __attribute__((ext_vector_type(2))) float v2f;
typedef __attribute__((ext_vector_type(8))) float v8f;

#define EPS 1e-8f

// ---------------------------------------------------------------------------
// Kernel 1: build per-edge data in WMMA-friendly packed layout.
//  For tile t (16 edges), packed[t*32 + m]      = (x[e], y[e])   (lanes 0-15)
//                         packed[t*32 + 16 + m] = (z[e], 0)      (lanes 16-31)
//  This single layout serves as BOTH the A-frag (16x4) and B-frag (4x16) of
//  V_WMMA_F32_16X16X4_F32 (K padded from 3 to 4 with zero).
// ---------------------------------------------------------------------------
__global__ void prep_kernel(const float* __restrict__ vertices,
                            const long long* __restrict__ faces,
                            int F, int E,
                            v2f* __restrict__ pc, v2f* __restrict__ pv,
                            float* __restrict__ normC, float* __restrict__ normV,
                            float* __restrict__ rowcnt)
{
    int e = blockIdx.x * blockDim.x + threadIdx.x;
    if (e >= E) return;

    int f, ia, ib;
    if (e < F)        { f = e;         ia = (int)faces[3*f+0]; ib = (int)faces[3*f+1]; }
    else if (e < 2*F) { f = e - F;     ia = (int)faces[3*f+1]; ib = (int)faces[3*f+2]; }
    else              { f = e - 2*F;   ia = (int)faces[3*f+2]; ib = (int)faces[3*f+0]; }

    float ax = vertices[3*ia+0], ay = vertices[3*ia+1], az = vertices[3*ia+2];
    float bx = vertices[3*ib+0], by = vertices[3*ib+1], bz = vertices[3*ib+2];

    float cx = 0.5f*(ax+bx), cy = 0.5f*(ay+by), cz = 0.5f*(az+bz);
    float vx = bx-ax+EPS,    vy = by-ay+EPS,    vz = bz-az+EPS;

    int t = e >> 4, m = e & 15;
    v2f cl = {cx, cy}; v2f ch = {cz, 0.f};
    v2f vl = {vx, vy}; v2f vh = {vz, 0.f};
    pc[t*32 + m]      = cl;
    pc[t*32 + 16 + m] = ch;
    pv[t*32 + m]      = vl;
    pv[t*32 + 16 + m] = vh;

    normC[e] = cx*cx + cy*cy + cz*cz;
    normV[e] = vx*vx + vy*vy + vz*vz;
    rowcnt[e] = 0.f;
}

// ---------------------------------------------------------------------------
// Kernel 2: all-pairs tile engine.  One wave32 owns one 16-row tile and one
// column slice; per column tile it issues two v_wmma_f32_16x16x4_f32 Gram
// products (centroids + directions) and thresholds the 16x16 result.
//  d^2        = |c_i|^2 + |c_j|^2 - 2 c_i.c_j        < (1+1e-6)^2
//  |cx norm|^2= |v_i|^2 |v_j|^2 - (v_i.v_j)^2        > (1e-5 - EPS)^2
// Symmetric condition => row-sum over all j != i equals row_cnt + col_cnt.
// ---------------------------------------------------------------------------
__global__ __launch_bounds__(256) void pair_kernel(
    const v2f*  __restrict__ pc, const v2f* __restrict__ pv,
    const float* __restrict__ normC, const float* __restrict__ normV,
    float* __restrict__ rowcnt, int NT, int NSLICES)
{
    const int lane  = threadIdx.x & 31;
    const int wave  = blockIdx.x * (blockDim.x >> 5) + (threadIdx.x >> 5);
    if (wave >= NT * NSLICES) return;          // wave-uniform: EXEC stays all-1s

    const int rt    = wave / NSLICES;          // row tile
    const int slice = wave % NSLICES;          // column slice
    const int half  = lane >> 4;
    const int nl    = lane & 15;
    const int mbase = rt * 16 + 8 * half;      // first of the 8 rows this lane's D-VGPRs cover

    // A fragments (held for the whole column loop)
    v2f a_c = pc[rt*32 + lane];
    v2f a_v = pv[rt*32 + lane];

    float rn_c[8], rn_v[8];
#pragma unroll
    for (int p = 0; p < 8; ++p) { rn_c[p] = normC[mbase + p]; rn_v[p] = normV[mbase + p]; }

    const float T1 = 1.000001f * 1.000001f;    // dist^2 threshold
    const float T2 = 9.99e-6f * 9.99e-6f;      // (1e-5 - EPS)^2 cross-norm^2 threshold

    float cnt[8];
#pragma unroll
    for (int p = 0; p < 8; ++p) cnt[p] = 0.f;

    for (int ct = slice; ct < NT; ct += NSLICES) {
        v2f b_c = pc[ct*32 + lane];
        v2f b_v = pv[ct*32 + lane];
        const int colg = ct*16 + nl;
        float cn_c = normC[colg];
        float cn_v = normV[colg];

        v8f zero = {};
        // D = A x B + 0 : 256 three-dim dot products per instruction
        v8f dotC = __builtin_amdgcn_wmma_f32_16x16x4_f32(
            false, a_c, false, b_c, (short)0, zero, false, false);
        v8f dotV = __builtin_amdgcn_wmma_f32_16x16x4_f32(
            false, a_v, false, b_v, (short)0, zero, false, false);

#pragma unroll
        for (int p = 0; p < 8; ++p) {
            float d2 = rn_c[p] + cn_c - 2.0f * dotC[p];
            float dv = dotV[p];
            float s  = rn_v[p] * cn_v - dv * dv;
            bool ok  = (d2 < T1) && (s > T2) && ((mbase + p) != colg); // exclude diagonal
            cnt[p] += ok ? 1.0f : 0.0f;
        }
    }

    // Reduce across each 16-lane half (counts are small integers in f32:
    // addition is exact and order-independent => deterministic).
#pragma unroll
    for (int msk = 1; msk <= 8; msk <<= 1) {
#pragma unroll
        for (int p = 0; p < 8; ++p)
            cnt[p] += __shfl_xor(cnt[p], msk, 32);
    }
    if (nl == 0) {
#pragma unroll
        for (int p = 0; p < 8; ++p)
            atomicAdd(&rowcnt[mbase + p], cnt[p]);
    }
}

// ---------------------------------------------------------------------------
// Kernel 3: segment-sum edges->faces (edge_to_face = e/3 in the reference),
// clip, dot with probs, mean.  One block, fixed-order tree reduction.
// ---------------------------------------------------------------------------
__global__ void finalize_kernel(const float* __restrict__ rowcnt,
                                const float* __restrict__ probs,
                                float* __restrict__ out, int F)
{
    __shared__ float red[1024];
    int tid = threadIdx.x;
    float acc = 0.f;
    for (int f = tid; f < F; f += 1024) {
        float cc = rowcnt[3*f] + rowcnt[3*f+1] + rowcnt[3*f+2];
        cc = fminf(fmaxf(cc, 0.f), 100.f);
        acc += probs[f] * cc;
    }
    red[tid] = acc;
    __syncthreads();
    for (int s = 512; s > 0; s >>= 1) {
        if (tid < s) red[tid] += red[tid + s];
        __syncthreads();
    }
    if (tid == 0) out[0] = red[0] / (float)F;
}

// ---------------------------------------------------------------------------
extern "C" void kernel_launch(void* const* d_in, const int* in_sizes, int n_in,
                              void* d_out, int out_size, void* d_ws, size_t ws_size,
                              hipStream_t stream)
{
    const float*     vertices = (const float*)d_in[0];
    const long long* faces    = (const long long*)d_in[1];
    const float*     probs    = (const float*)d_in[2];
    float*           out      = (float*)d_out;

    const int F  = in_sizes[2];     // 8192
    const int E  = 3 * F;           // 24576, multiple of 16
    const int NT = E / 16;          // 1536 tiles
    const int NSLICES = 16;

    char* ws = (char*)d_ws;
    size_t off = 0;
    v2f*   pc     = (v2f*)(ws + off);   off += (size_t)NT * 32 * sizeof(v2f);
    v2f*   pv     = (v2f*)(ws + off);   off += (size_t)NT * 32 * sizeof(v2f);
    float* normC  = (float*)(ws + off); off += (size_t)E * sizeof(float);
    float* normV  = (float*)(ws + off); off += (size_t)E * sizeof(float);
    float* rowcnt = (float*)(ws + off); off += (size_t)E * sizeof(float);
    (void)ws_size; (void)n_in; (void)out_size;  // ~1.06 MB total scratch

    prep_kernel<<<(E + 255) / 256, 256, 0, stream>>>(
        vertices, faces, F, E, pc, pv, normC, normV, rowcnt);

    const int totalWaves = NT * NSLICES;            // 24576 waves
    const int blocks = (totalWaves * 32 + 255) / 256;
    pair_kernel<<<blocks, 256, 0, stream>>>(pc, pv, normC, normV, rowcnt, NT, NSLICES);

    finalize_kernel<<<1, 1024, 0, stream>>>(rowcnt, probs, out, F);
}